// LenetOptNet_11690900979742
// MI455X (gfx1250) — compile-verified
//
#include <hip/hip_runtime.h>
#include <hip/hip_bf16.h>
#include <math.h>

// -------- CDNA5 WMMA types: f32 16x16x4 => A/B = 2 VGPRs, C/D = 8 VGPRs -----
typedef float v2f __attribute__((ext_vector_type(2)));
typedef float v8f __attribute__((ext_vector_type(8)));

// ---------------------------------------------------------------------------
// Cooperative 64x64 Cholesky in LDS (lower, in place). 256 threads.
// Padding rows/cols (>=50) are identity so the factorization is exact.
// ---------------------------------------------------------------------------
__device__ __forceinline__ void block_cholesky64(float (*H)[65], int t) {
  for (int k = 0; k < 64; ++k) {
    if (t == 0) H[k][k] = sqrtf(fmaxf(H[k][k], 1e-30f));
    __syncthreads();
    if (t > k && t < 64) H[t][k] /= H[k][k];
    __syncthreads();
    // trailing update: writes hit distinct (i,j) with j>k; reads only column k
    for (int j = k + 1; j < 64; ++j) {
      int i = j + t;
      if (i < 64) H[i][j] -= H[i][k] * H[j][k];
    }
    __syncthreads();
  }
}

// x <- (L L^T)^{-1} x  with L in lower triangle of H. 256 threads.
__device__ __forceinline__ void chol_solve64(const float (*H)[65], float* x, int t) {
  for (int k = 0; k < 64; ++k) {              // forward: L y = b
    if (t == 0) x[k] /= H[k][k];
    __syncthreads();
    if (t > k && t < 64) x[t] -= H[t][k] * x[k];
    __syncthreads();
  }
  for (int k = 63; k >= 0; --k) {             // backward: L^T x = y
    if (t == 0) x[k] /= H[k][k];
    __syncthreads();
    if (t < k) x[t] -= H[k][t] * x[k];
    __syncthreads();
  }
}

// Tree reduction of red[0..63]; all 256 threads call; result broadcast.
__device__ __forceinline__ float block_reduce64(float* red, int t, bool domin) {
  __syncthreads();
  for (int off = 32; off > 0; off >>= 1) {
    if (t < off) red[t] = domin ? fminf(red[t], red[t + off]) : (red[t] + red[t + off]);
    __syncthreads();
  }
  float r = red[0];
  __syncthreads();
  return r;
}

// ---------------------------------------------------------------------------
// conv1 (1->20, 5x5, VALID) + 2x2 maxpool : x[512,1,28,28] -> f1[512,20,12,12]
// ---------------------------------------------------------------------------
__global__ void k_conv1(const float* __restrict__ x, const float* __restrict__ w,
                        const float* __restrict__ bias, float* __restrict__ f1) {
  int idx = blockIdx.x * 256 + threadIdx.x;
  if (idx >= 512 * 20 * 144) return;
  int ox = idx % 12, oy = (idx / 12) % 12, c = (idx / 144) % 20, b = idx / 2880;
  const float* xb = x + b * 784;
  const float* wc = w + c * 25;
  float mx = -3.4e38f;
  for (int dy = 0; dy < 2; ++dy)
    for (int dx = 0; dx < 2; ++dx) {
      int py = 2 * oy + dy, px = 2 * ox + dx;
      float acc = bias[c];
      for (int ky = 0; ky < 5; ++ky)
        for (int kx = 0; kx < 5; ++kx)
          acc += xb[(py + ky) * 28 + (px + kx)] * wc[ky * 5 + kx];
      mx = fmaxf(mx, acc);
    }
  f1[idx] = mx;
}

// ---------------------------------------------------------------------------
// conv2 (20->50, 5x5) + pool : f1[b,20,12,12] (LDS-staged) -> feat[512,800]
// feat layout matches reshape(B,-1) of [B,50,4,4]: o = c*16 + oy*4 + ox
// ---------------------------------------------------------------------------
__global__ void k_conv2(const float* __restrict__ f1, const float* __restrict__ w,
                        const float* __restrict__ bias, float* __restrict__ feat) {
  __shared__ float f1s[2880];
  int t = threadIdx.x, b = blockIdx.x;
  for (int i = t; i < 2880; i += 256) f1s[i] = f1[b * 2880 + i];
  __syncthreads();
  for (int o = t; o < 800; o += 256) {
    int c = o >> 4, pos = o & 15, oy = pos >> 2, ox = pos & 3;
    const float* wc = w + c * 500;
    float mx = -3.4e38f;
    for (int dy = 0; dy < 2; ++dy)
      for (int dx = 0; dx < 2; ++dx) {
        int py = 2 * oy + dy, px = 2 * ox + dx;
        float acc = bias[c];
        for (int ci = 0; ci < 20; ++ci)
          for (int ky = 0; ky < 5; ++ky)
            for (int kx = 0; kx < 5; ++kx)
              acc += f1s[(ci * 12 + py + ky) * 12 + (px + kx)] * wc[(ci * 5 + ky) * 5 + kx];
        mx = fmaxf(mx, acc);
      }
    feat[b * 800 + o] = mx;
  }
}

// ---------------------------------------------------------------------------
// Build concatenated head weights Wcat[304][800] (rows: z0 0..49 | s0 50..249
// | p 250..299 | zero pad 300..303) and bcat[304].
// ---------------------------------------------------------------------------
__global__ void k_prep_w(const float* __restrict__ zw, const float* __restrict__ zb,
                         const float* __restrict__ sw, const float* __restrict__ sb,
                         const float* __restrict__ ow, const float* __restrict__ ob,
                         float* __restrict__ Wcat, float* __restrict__ bcat) {
  int r = blockIdx.x, t = threadIdx.x;
  const float* src = nullptr;
  float bv = 0.f;
  if (r < 50)       { src = zw + r * 800;        bv = zb[r]; }
  else if (r < 250) { src = sw + (r - 50) * 800; bv = sb[r - 50]; }
  else if (r < 300) { src = ow + (r - 250) * 800; bv = ob[r - 250]; }
  for (int k = t; k < 800; k += 256) Wcat[r * 800 + k] = src ? src[k] : 0.f;
  if (t == 0) bcat[r] = bv;
}

// ---------------------------------------------------------------------------
// Build Qp[64][64] = tril(L) tril(L)^T + eps*I, identity-padded; Gp[200][64]
// zero-padded; cholQ[64][64] = chol(Qp) for the z-init solve.
// ---------------------------------------------------------------------------
__global__ void k_prep_qg(const float* __restrict__ L, const float* __restrict__ G,
                          float* __restrict__ Qp, float* __restrict__ Gp,
                          float* __restrict__ cholQ) {
  __shared__ float Hs[64][65];
  int t = threadIdx.x;
  for (int idx = t; idx < 4096; idx += 256) {
    int i = idx >> 6, j = idx & 63;
    float v;
    if (i < 50 && j < 50) {
      v = (i == j) ? 1e-4f : 0.f;
      int km = i < j ? i : j;
      for (int k = 0; k <= km; ++k) v += L[i * 50 + k] * L[j * 50 + k];  // tril
    } else {
      v = (i == j) ? 1.f : 0.f;   // identity padding keeps chol/solve exact
    }
    Qp[idx] = v;
    Hs[i][j] = v;
  }
  for (int idx = t; idx < 200 * 64; idx += 256) {
    int n = idx & 63;
    Gp[idx] = (n < 50) ? G[(idx >> 6) * 50 + n] : 0.f;
  }
  __syncthreads();
  block_cholesky64(Hs, t);
  for (int idx = t; idx < 4096; idx += 256) cholQ[idx] = Hs[idx >> 6][idx & 63];
}

// ---------------------------------------------------------------------------
// Linear heads as one WMMA GEMM: zsp[512,300] = feat[512,800] @ Wcat^T + bcat.
// One wave per 16x16 output tile, K-loop of v_wmma_f32_16x16x4_f32.
// A layout (16x4 f32): M = lane%16, K = 2*(lane/16) + vgpr.
// B layout (4x16 f32): N = lane%16, K = 2*(lane/16) + vgpr.
// C layout: M = r + 8*(lane/16), N = lane%16.
// ---------------------------------------------------------------------------
__global__ void __launch_bounds__(128) k_linear(const float* __restrict__ feat,
                                                const float* __restrict__ Wcat,
                                                const float* __restrict__ bcat,
                                                float* __restrict__ zsp) {
  int t = threadIdx.x;
  int wv = t >> 5, lane = t & 31, lh = lane >> 4, lm = lane & 15;
  int ntile = blockIdx.y * 4 + wv;
  if (ntile >= 19) return;                      // wave-uniform exit
  int mbase = blockIdx.x * 16, nbase = ntile * 16;
  const float* arow = feat + (mbase + lm) * 800;   // A: feat row per lane
  const float* brow = Wcat + (nbase + lm) * 800;   // B[k][n] = Wcat[n][k]
  __builtin_prefetch(arow, 0, 3);                  // global_prefetch_b8
  __builtin_prefetch(brow, 0, 3);
  v8f acc = {};
  for (int kb = 0; kb < 800; kb += 4) {
    int k0 = kb + 2 * lh;
    v2f A, B;
    A.x = arow[k0]; A.y = arow[k0 + 1];
    B.x = brow[k0]; B.y = brow[k0 + 1];
    acc = __builtin_amdgcn_wmma_f32_16x16x4_f32(false, A, false, B, (short)0, acc,
                                                false, false);
  }
  for (int r = 0; r < 8; ++r) {
    int row = mbase + r + 8 * lh;
    int col = nbase + lm;
    if (col < 300) zsp[row * 300 + col] = acc[r] + bcat[col];
  }
}

// ---------------------------------------------------------------------------
// Batched PDIPM QP solve: one 256-thread block (8 wave32) per batch element.
// G (200x65, conflict-padded) and Q (64x65) are staged in LDS ONCE and reused
// for all 20 iterations: every WMMA operand fetch and every matvec is an
// LDS read (stride 65 => bank (k+i)%64, conflict-free), eliminating ~7 GB of
// repeated L2 traffic. H = Qp + G^T diag(lam/s) G built with
// v_wmma_f32_16x16x4_f32 (4x4 tile grid over padded 64x64, 2 tiles/wave).
// LDS footprint ~91 KB/WG -> 3 WGs (24 wave32) per 320 KB WGP.
// ---------------------------------------------------------------------------
__global__ void __launch_bounds__(256) k_qp(const float* __restrict__ zsp,
                                            const float* __restrict__ Qp,
                                            const float* __restrict__ cholQ,
                                            const float* __restrict__ Gp,
                                            float* __restrict__ out) {
  __shared__ float Hs[64][65];
  __shared__ float Qs[64][65];
  __shared__ float Gls[200][65];
  __shared__ float zs[64], ps[64], rzs[64], dzs[64], z0s[64], red[64];
  __shared__ float hs[200], ss[200], lams[200], Ds[200], rps[200];
  __shared__ float dsa[200], dlama[200], rcs[200];
  const int t = threadIdx.x;
  const int b = blockIdx.x;
  const float* zrow = zsp + b * 300;

  if (t < 64) {
    z0s[t] = (t < 50) ? zrow[t] : 0.f;
    ps[t]  = (t < 50) ? zrow[250 + t] : 0.f;
  }
  // one-time LDS staging: chol(Q) -> Hs (for z-init solve), Q -> Qs, G -> Gls
  for (int idx = t; idx < 4096; idx += 256) {
    Hs[idx >> 6][idx & 63] = cholQ[idx];
    Qs[idx >> 6][idx & 63] = Qp[idx];
  }
  for (int idx = t; idx < 12800; idx += 256)
    Gls[idx >> 6][idx & 63] = Gp[idx];
  __syncthreads();
  // h = G z0 + s0
  for (int m = t; m < 200; m += 256) {
    float acc = zrow[50 + m];
    for (int n = 0; n < 50; ++n) acc += Gls[m][n] * z0s[n];
    hs[m] = acc;
  }
  if (t < 64) zs[t] = -ps[t];
  __syncthreads();
  chol_solve64(Hs, zs, t);                      // z = Q^{-1}(-p), pad stays 0
  for (int m = t; m < 200; m += 256) {
    float acc = 0.f;
    for (int n = 0; n < 50; ++n) acc += Gls[m][n] * zs[n];
    ss[m] = fmaxf(hs[m] - acc, 1.0f);
    lams[m] = 1.0f;
  }
  __syncthreads();

  for (int it = 0; it < 20; ++it) {
    // rp = G z + s - h ; D = lam/s
    for (int m = t; m < 200; m += 256) {
      float acc = 0.f;
      for (int n = 0; n < 50; ++n) acc += Gls[m][n] * zs[n];
      rps[m] = acc + ss[m] - hs[m];
      Ds[m] = lams[m] / ss[m];
    }
    // rz = Q z + p + G^T lam (pad rows evaluate to 0)
    if (t < 64) {
      float acc = ps[t];
      for (int j = 0; j < 64; ++j) acc += Qs[t][j] * zs[j];
      for (int m = 0; m < 200; ++m) acc += Gls[m][t] * lams[m];
      rzs[t] = acc;
    }
    if (t < 64) {
      float a = 0.f;
      for (int m = t; m < 200; m += 64) a += ss[m] * lams[m];
      red[t] = a;
    }
    float mu = block_reduce64(red, t, false) * 0.005f;   // mean over 200

    // ---- H = Qs + (D*G)^T G via WMMA f32 16x16x4, K = 200, LDS operands ----
    {
      int wv = t >> 5, lane = t & 31, lh = lane >> 4, lm = lane & 15;
      for (int tt = wv; tt < 16; tt += 8) {
        int ti = tt >> 2, tj = tt & 3;
        const int ib = ti * 16 + lm, jb = tj * 16 + lm;
        v8f acc = {};
        for (int kb = 0; kb < 200; kb += 4) {
          int k0 = kb + 2 * lh;
          v2f A, B;
          A.x = Ds[k0]     * Gls[k0][ib];          // A[i,k] = D[k]*G[k,i]
          A.y = Ds[k0 + 1] * Gls[k0 + 1][ib];
          B.x = Gls[k0][jb];                       // B[k,j] = G[k,j]
          B.y = Gls[k0 + 1][jb];
          acc = __builtin_amdgcn_wmma_f32_16x16x4_f32(false, A, false, B, (short)0,
                                                      acc, false, false);
        }
        for (int r = 0; r < 8; ++r) {
          int i = ti * 16 + r + 8 * lh;
          Hs[i][jb] = acc[r] + Qs[i][jb];
        }
      }
    }
    __syncthreads();
    block_cholesky64(Hs, t);

    // predictor: rhs_a = -(rz + G^T(D*rp - lam))
    if (t < 64) {
      float acc = rzs[t];
      for (int m = 0; m < 200; ++m) acc += Gls[m][t] * (Ds[m] * rps[m] - lams[m]);
      dzs[t] = -acc;
    }
    __syncthreads();
    chol_solve64(Hs, dzs, t);
    for (int m = t; m < 200; m += 256) {
      float gd = 0.f;
      for (int n = 0; n < 50; ++n) gd += Gls[m][n] * dzs[n];
      float dl = Ds[m] * (gd + rps[m]) - lams[m];
      dlama[m] = dl;
      dsa[m] = -(ss[m] * lams[m] + ss[m] * dl) / lams[m];
    }
    __syncthreads();
    if (t < 64) {
      float r1 = 3.4e38f;
      for (int m = t; m < 200; m += 64) {
        if (dsa[m]   < 0.f) r1 = fminf(r1, -ss[m]   / dsa[m]);
        if (dlama[m] < 0.f) r1 = fminf(r1, -lams[m] / dlama[m]);
      }
      red[t] = r1;
    }
    float a_a = fminf(1.f, 0.99f * block_reduce64(red, t, true));
    if (t < 64) {
      float acc = 0.f;
      for (int m = t; m < 200; m += 64)
        acc += (ss[m] + a_a * dsa[m]) * (lams[m] + a_a * dlama[m]);
      red[t] = acc;
    }
    float mu_a = block_reduce64(red, t, false) * 0.005f;
    float sig = mu_a / mu;
    sig = sig * sig * sig;

    // corrector
    for (int m = t; m < 200; m += 256)
      rcs[m] = ss[m] * lams[m] + dsa[m] * dlama[m] - sig * mu;
    __syncthreads();
    if (t < 64) {
      float acc = rzs[t];
      for (int m = 0; m < 200; ++m)
        acc += Gls[m][t] * (Ds[m] * rps[m] - rcs[m] / ss[m]);
      dzs[t] = -acc;
    }
    __syncthreads();
    chol_solve64(Hs, dzs, t);
    for (int m = t; m < 200; m += 256) {
      float gd = 0.f;
      for (int n = 0; n < 50; ++n) gd += Gls[m][n] * dzs[n];
      float dl = Ds[m] * (gd + rps[m]) - rcs[m] / ss[m];
      dlama[m] = dl;
      dsa[m] = -(rcs[m] + ss[m] * dl) / lams[m];
    }
    __syncthreads();
    if (t < 64) {
      float r1 = 3.4e38f;
      for (int m = t; m < 200; m += 64) {
        if (dsa[m]   < 0.f) r1 = fminf(r1, -ss[m]   / dsa[m]);
        if (dlama[m] < 0.f) r1 = fminf(r1, -lams[m] / dlama[m]);
      }
      red[t] = r1;
    }
    float a = fminf(1.f, 0.99f * block_reduce64(red, t, true));
    if (t < 64) zs[t] += a * dzs[t];
    for (int m = t; m < 200; m += 256) {
      ss[m]   += a * dsa[m];
      lams[m] += a * dlama[m];
    }
    __syncthreads();
  }

  // log_softmax over z[0:10]
  if (t == 0) {
    float mx = -3.4e38f;
    for (int i = 0; i < 10; ++i) mx = fmaxf(mx, zs[i]);
    float se = 0.f;
    for (int i = 0; i < 10; ++i) se += expf(zs[i] - mx);
    red[0] = mx + logf(se);
  }
  __syncthreads();
  if (t < 10) out[b * 10 + t] = zs[t] - red[0];
}

// ---------------------------------------------------------------------------
extern "C" void kernel_launch(void* const* d_in, const int* in_sizes, int n_in,
                              void* d_out, int out_size, void* d_ws, size_t ws_size,
                              hipStream_t stream) {
  const float* x   = (const float*)d_in[0];
  const float* c1w = (const float*)d_in[1];
  const float* c1b = (const float*)d_in[2];
  const float* c2w = (const float*)d_in[3];
  const float* c2b = (const float*)d_in[4];
  const float* ow  = (const float*)d_in[5];
  const float* ob  = (const float*)d_in[6];
  const float* zw  = (const float*)d_in[7];
  const float* zb  = (const float*)d_in[8];
  const float* sw  = (const float*)d_in[9];
  const float* sb  = (const float*)d_in[10];
  const float* L   = (const float*)d_in[11];
  const float* G   = (const float*)d_in[12];

  float* ws = (float*)d_ws;
  size_t o = 0;
  float* f1    = ws + o; o += (size_t)512 * 2880;   // conv1+pool output
  float* feat  = ws + o; o += (size_t)512 * 800;    // conv2+pool output
  float* zsp   = ws + o; o += (size_t)512 * 300;    // [z0|s0|p] per batch
  float* Qp    = ws + o; o += 4096;                 // padded Q (64x64)
  float* cholQ = ws + o; o += 4096;                 // chol(Qp)
  float* Gp    = ws + o; o += 200 * 64;             // padded G
  float* Wcat  = ws + o; o += (size_t)304 * 800;    // concat head weights
  float* bcat  = ws + o; o += 304;                  // concat head biases
  float* out = (float*)d_out;

  k_prep_w <<<dim3(304),     dim3(256), 0, stream>>>(zw, zb, sw, sb, ow, ob, Wcat, bcat);
  k_prep_qg<<<dim3(1),       dim3(256), 0, stream>>>(L, G, Qp, Gp, cholQ);
  k_conv1  <<<dim3(5760),    dim3(256), 0, stream>>>(x, c1w, c1b, f1);
  k_conv2  <<<dim3(512),     dim3(256), 0, stream>>>(f1, c2w, c2b, feat);
  k_linear <<<dim3(32, 5),   dim3(128), 0, stream>>>(feat, Wcat, bcat, zsp);
  k_qp     <<<dim3(512),     dim3(256), 0, stream>>>(zsp, Qp, cholQ, Gp, out);
}